// LearnableSamples_64536178590319
// MI455X (gfx1250) — compile-verified
//
#include <hip/hip_runtime.h>

typedef __attribute__((ext_vector_type(2))) float v2f;
typedef __attribute__((ext_vector_type(8))) float v8f;

// Problem constants (match the reference)
#define PM 2048
#define PN 1024
#define PD 64
#define SQRT_DT 0.03125f          // sqrt(1/1024)
#define SIGMA 0.125f              // 1/sqrt(64)
#define SIG_SQDT (SIGMA * SQRT_DT)

// One wave per (m, 16-column group) strip. Each strip walks 64 tiles of
// 16 timesteps, doing the inclusive prefix-sum with a lower-triangular
// 16x16 ones-matrix via chained V_WMMA_F32_16X16X4_F32 (K in 4 slices).
//
// WMMA f32 operand layouts (ISA 7.12.2, wave32):
//   A (16x4):  lane%16 = output row M; lanes<16 hold K={4k,4k+1} in v[0],v[1];
//              lanes>=16 hold K={4k+2,4k+3}.
//   B (4x16):  lane%16 = output col N; same half-wave K split as A.
//   C/D (16x16): VGPR i holds rows {i, i+8} (half-wave select), col = lane%16.
__global__ void __launch_bounds__(256)
LearnableSamples_64536178590319_kernel(const float* __restrict__ x0,
                                       const float* __restrict__ theta,
                                       float* __restrict__ X,
                                       float* __restrict__ dW) {
    const int lane  = threadIdx.x & 31;
    const int wave  = threadIdx.x >> 5;
    const int strip = blockIdx.x * (blockDim.x >> 5) + wave;   // 0..M*4-1
    const int m     = strip >> 2;                              // row of x0
    const int cg    = strip & 3;                               // 16-col group
    const int col   = lane & 15;                               // N-col / A-row
    const int half  = lane >> 4;                               // K-pair select

    const int    gcol      = cg * 16 + col;
    const size_t thetaBase = (size_t)m * PN * PD;
    const size_t xBase     = (size_t)m * (PN + 1) * PD;

    const float x0v = x0[(size_t)m * PD + gcol];
    // X[m][0][:] = x0 (both half-waves write the same value; benign)
    X[xBase + gcol] = x0v;

    // running value at the end of the previous block: x0 + sigma*cumsum
    float base = x0v;

    for (int b = 0; b < PN / 16; ++b) {
        const int step0 = b * 16;

        // C seeded with the carried value, replicated across all 16 rows.
        v8f c;
#pragma unroll
        for (int i = 0; i < 8; ++i) c[i] = base;

#pragma unroll
        for (int k = 0; k < 4; ++k) {
            const int kc = 4 * k + 2 * half;        // this lane's K columns
            const size_t r0 = thetaBase + (size_t)(step0 + kc) * PD + gcol;
            const size_t r1 = r0 + PD;

            const float t0 = theta[r0];
            const float t1 = theta[r1];

            // dW = theta * sqrt(dt)
            dW[r0] = t0 * SQRT_DT;
            dW[r1] = t1 * SQRT_DT;

            // B chunk: rows K=kc,kc+1 of the tile, pre-scaled by sigma*sqrt(dt)
            v2f Bv;
            Bv[0] = t0 * SIG_SQDT;
            Bv[1] = t1 * SIG_SQDT;

            // A chunk: lower-triangular ones, columns kc, kc+1
            v2f Av;
            Av[0] = (col >= kc)     ? 1.0f : 0.0f;
            Av[1] = (col >= kc + 1) ? 1.0f : 0.0f;

            // c = A*B + c  ->  accumulates the inclusive prefix rows
            c = __builtin_amdgcn_wmma_f32_16x16x4_f32(
                    /*neg_a=*/false, Av, /*neg_b=*/false, Bv,
                    /*c_mod=*/(short)0, c,
                    /*reuse_a=*/false, /*reuse_b=*/false);
        }

        // Store X rows step0+1 .. step0+16 for this column group.
#pragma unroll
        for (int i = 0; i < 8; ++i) {
            const int step = step0 + (i + 8 * half) + 1;
            X[xBase + (size_t)step * PD + gcol] = c[i];
        }

        // Carry = last prefix row (row 15) = VGPR7 of lanes 16..31.
        base = __shfl(c[7], 16 + col, 32);
    }
}

extern "C" void kernel_launch(void* const* d_in, const int* in_sizes, int n_in,
                              void* d_out, int out_size, void* d_ws, size_t ws_size,
                              hipStream_t stream) {
    (void)in_sizes; (void)n_in; (void)d_ws; (void)ws_size; (void)out_size;

    const float* x0    = (const float*)d_in[0];
    // d_in[1] = t_grid: unused by the reference outputs
    const float* theta = (const float*)d_in[2];

    float* X  = (float*)d_out;                                  // [M, N+1, D]
    float* dW = (float*)d_out + (size_t)PM * (PN + 1) * PD;     // [M, N, D]

    const int strips         = PM * 4;      // 8192 waves
    const int wavesPerBlock  = 8;
    dim3 block(32 * wavesPerBlock);
    dim3 grid(strips / wavesPerBlock);      // 1024 blocks

    LearnableSamples_64536178590319_kernel<<<grid, block, 0, stream>>>(
        x0, theta, X, dW);
}